// kmeans_4097398800606
// MI455X (gfx1250) — compile-verified
//
#include <hip/hip_runtime.h>

// ---------------------------------------------------------------------------
// K-means (N=1M, K=64, NC=64, 5 iters) for gfx1250 / MI455X.
//   - distance GEMM via V_WMMA_F32_16X16X4_F32 (fp32-exact vs reference)
//   - X tiles staged to LDS in B-fragment order via GLOBAL_LOAD_ASYNC_TO_LDS_B64
//     (ASYNCcnt-tracked, double-buffered per wave -> global latency hidden)
//   - argmin via per-lane reduce + __shfl_xor lane-half merge
//   - scatter-mean via LDS ds_add_f32 partials + global atomic flush
// ---------------------------------------------------------------------------

typedef float v2f __attribute__((ext_vector_type(2)));
typedef float v8f __attribute__((ext_vector_type(8)));
typedef int   v2i_vs __attribute__((vector_size(8)));   // exact builtin pointee type

#define N_PTS        (1u << 20)
#define KDIM         64
#define NC           64
#define NITER        5
#define THREADS      256
#define WAVES        8
#define PTS_PER_BLK  2048
#define PTS_PER_WAVE (PTS_PER_BLK / WAVES)   // 256
#define CHUNKS       (PTS_PER_WAVE / 16)     // 16 chunks of 16 points
#define PART_STRIDE  65                      // pad: bank = (c + d) & 63

// ---- CDNA5 async global->LDS path (guarded; sync fallback keeps structure) ----
#if defined(__has_builtin)
#if __has_builtin(__builtin_amdgcn_global_load_async_to_lds_b64) && \
    __has_builtin(__builtin_amdgcn_s_wait_asynccnt)
#define USE_ASYNC_LDS 1
#endif
#endif

#ifdef USE_ASYNC_LDS
// signature (probe-confirmed): (v2i AS1*, v2i AS3*, imm int offset, imm int cpol)
#define ASYNC_CP_B64(gptr, lptr)                                               \
  __builtin_amdgcn_global_load_async_to_lds_b64(                               \
      (__attribute__((address_space(1))) v2i_vs*)(uintptr_t)(gptr),            \
      (__attribute__((address_space(3))) v2i_vs*)(unsigned int)(uintptr_t)(lptr), \
      0, 0)
#define WAIT_ASYNC(n) __builtin_amdgcn_s_wait_asynccnt(n)
#else
#define ASYNC_CP_B64(gptr, lptr)                                               \
  (*(v2f*)(lptr) = *(const v2f*)(gptr))
#define WAIT_ASYNC(n) ((void)0)
#endif

__global__ __launch_bounds__(THREADS) void kmeans_zero(float* __restrict__ sums,
                                                       float* __restrict__ counts) {
  int i = blockIdx.x * THREADS + threadIdx.x;
  if (i < NC * KDIM) sums[i] = 0.0f;
  if (i < NC) counts[i] = 0.0f;
}

__global__ __launch_bounds__(THREADS) void kmeans_assign_acc(
    const float* __restrict__ X, const float* __restrict__ mu,
    float* __restrict__ sums, float* __restrict__ counts) {
  // mu A-fragments pre-swizzled to WMMA layout: [tile t][kstep s][lane]
  __shared__ v2f   s_muA[4 * 16 * 32];          // 16 KB
  // per-wave double-buffered X tile in B-fragment order: [wave][buf][kstep*32+lane]
  __shared__ v2f   s_xb[WAVES][2][16 * 32];     // 64 KB
  __shared__ float s_mu2[NC];
  __shared__ float s_part[NC * PART_STRIDE];    // 16.25 KB padded partial sums
  __shared__ float s_cnt[NC];

  const int tid = threadIdx.x;

  // ---- block init: zero partials, mu2, stage A fragments ----
  for (int i = tid; i < NC * PART_STRIDE; i += THREADS) s_part[i] = 0.0f;
  if (tid < NC) {
    s_cnt[tid] = 0.0f;
    const float* mrow = mu + tid * KDIM;
    float m2 = 0.0f;
#pragma unroll
    for (int k = 0; k < KDIM; ++k) m2 = __builtin_fmaf(mrow[k], mrow[k], m2);
    s_mu2[tid] = m2;
  }
  // A-matrix 16x4 f32 layout: lane l holds M=l&15; VGPR0 = K {0 | 2}, VGPR1 = K {1 | 3}
  for (int i = tid; i < 4 * 16 * 32; i += THREADS) {
    int t = i >> 9;
    int s = (i >> 5) & 15;
    int l = i & 31;
    int M  = t * 16 + (l & 15);
    int k0 = 4 * s + ((l >> 4) << 1);
    v2f a;
    a.x = mu[M * KDIM + k0];
    a.y = mu[M * KDIM + k0 + 1];
    s_muA[i] = a;
  }
  __syncthreads();

  const int wave = tid >> 5;
  const int lane = tid & 31;
  const int hi   = lane >> 4;       // lane-half: 0 -> M=r / K{0,1}, 1 -> M=r+8 / K{2,3}
  const int lp   = lane & 15;       // point slot within 16-point chunk
  const size_t base_pt = (size_t)blockIdx.x * PTS_PER_BLK + (size_t)wave * PTS_PER_WAVE;

  // issue the 16 async B-fragment copies for one 16-point chunk (swizzling copy:
  // each lane picks its own global source and LDS fragment destination)
  auto issue_tile = [&](int chunk, int buf) {
    const float* xrow = X + (base_pt + (size_t)chunk * 16 + lp) * KDIM;
    v2f* dst = &s_xb[wave][buf][0];
#pragma unroll
    for (int s = 0; s < 16; ++s) {
      ASYNC_CP_B64(xrow + 4 * s + 2 * hi, &dst[s * 32 + lane]);
    }
  };

  const v8f vzero = {};

  // prologue: prefetch chunk 0
  issue_tile(0, 0);

  for (int chunk = 0; chunk < CHUNKS; ++chunk) {
    const int cur = chunk & 1;
    if (chunk + 1 < CHUNKS) {
      issue_tile(chunk + 1, cur ^ 1);   // async fill of the other buffer
      WAIT_ASYNC(16);                   // retire current tile (16 older copies)
    } else {
      WAIT_ASYNC(0);
    }
    asm volatile("" ::: "memory");      // keep LDS reads below the wait

    // ---- D = mu_tile x X_chunk^T  (4 cluster tiles x 16 k-steps) ----
    v8f acc[4];
#pragma unroll
    for (int t = 0; t < 4; ++t) acc[t] = vzero;

    const v2f* bsrc = &s_xb[wave][cur][0];
#pragma unroll
    for (int s = 0; s < 16; ++s) {
      v2f b = bsrc[s * 32 + lane];      // lane-contiguous ds_load_b64, conflict-free
#pragma unroll
      for (int t = 0; t < 4; ++t) {
        v2f a = s_muA[(t * 16 + s) * 32 + lane];
        acc[t] = __builtin_amdgcn_wmma_f32_16x16x4_f32(
            /*neg_a=*/false, a, /*neg_b=*/false, b,
            /*c_mod=*/(short)0, acc[t], /*reuse_a=*/false, /*reuse_b=*/false);
      }
    }

    // ---- per-point argmin of (mu2[c] - 2*dot) over 64 clusters ----
    float bestv = 3.4028235e38f;
    int   bestc = 0;
#pragma unroll
    for (int t = 0; t < 4; ++t) {
#pragma unroll
      for (int r = 0; r < 8; ++r) {
        int   c  = t * 16 + r + 8 * hi;   // C layout: VGPR r -> M=r (lo) / r+8 (hi)
        float sc = __builtin_fmaf(-2.0f, acc[t][r], s_mu2[c]);
        if (sc < bestv) { bestv = sc; bestc = c; }
      }
    }
    // merge the two lane halves (same point, disjoint cluster rows)
    float pv = __shfl_xor(bestv, 16, 32);
    int   pc = __shfl_xor(bestc, 16, 32);
    if (pv < bestv || (pv == bestv && pc < bestc)) { bestv = pv; bestc = pc; }

    // ---- scatter-accumulate into LDS partials (X re-read hits cache) ----
    const float* xrow = X + (base_pt + (size_t)chunk * 16 + lp) * KDIM;
    if (!hi) atomicAdd(&s_cnt[bestc], 1.0f);
    const float4* x4  = reinterpret_cast<const float4*>(xrow + 32 * hi);
    float*        dst = &s_part[bestc * PART_STRIDE + 32 * hi];
#pragma unroll
    for (int j = 0; j < 8; ++j) {
      float4 v = x4[j];
      atomicAdd(dst + 4 * j + 0, v.x);
      atomicAdd(dst + 4 * j + 1, v.y);
      atomicAdd(dst + 4 * j + 2, v.z);
      atomicAdd(dst + 4 * j + 3, v.w);
    }
  }
  __syncthreads();

  // ---- flush block partials to global ----
  for (int i = tid; i < NC * KDIM; i += THREADS) {
    int c = i >> 6, d = i & 63;
    atomicAdd(&sums[i], s_part[c * PART_STRIDE + d]);
  }
  if (tid < NC) atomicAdd(&counts[tid], s_cnt[tid]);
}

__global__ __launch_bounds__(THREADS) void kmeans_update(
    const float* __restrict__ sums, const float* __restrict__ counts,
    const float* __restrict__ mu_in, float* __restrict__ mu_out) {
  int i = blockIdx.x * THREADS + threadIdx.x;
  if (i >= NC * KDIM) return;
  float cnt = counts[i >> 6];
  mu_out[i] = (cnt > 0.0f) ? (sums[i] / cnt) : mu_in[i];
}

extern "C" void kernel_launch(void* const* d_in, const int* in_sizes, int n_in,
                              void* d_out, int out_size, void* d_ws, size_t ws_size,
                              hipStream_t stream) {
  const float* X   = (const float*)d_in[0];   // (N, 64) fp32
  const float* mu0 = (const float*)d_in[1];   // (64, 64) fp32
  (void)in_sizes; (void)n_in; (void)out_size; (void)ws_size;

  float* mu_buf = (float*)d_ws;               // 4096 floats
  float* sums   = mu_buf + NC * KDIM;         // 4096 floats
  float* counts = sums + NC * KDIM;           // 64 floats

  (void)hipMemcpyAsync(mu_buf, mu0, NC * KDIM * sizeof(float),
                       hipMemcpyDeviceToDevice, stream);

  const int nblk = N_PTS / PTS_PER_BLK;       // 512 blocks x 8 waves
  for (int it = 0; it < NITER; ++it) {
    kmeans_zero<<<(NC * KDIM + THREADS - 1) / THREADS, THREADS, 0, stream>>>(sums, counts);
    kmeans_assign_acc<<<nblk, THREADS, 0, stream>>>(X, mu_buf, sums, counts);
    float* outp = (it == NITER - 1) ? (float*)d_out : mu_buf;
    kmeans_update<<<(NC * KDIM + THREADS - 1) / THREADS, THREADS, 0, stream>>>(
        sums, counts, mu_buf, outp);
  }
}